// Prototype_18322330485494
// MI455X (gfx1250) — compile-verified
//
#include <hip/hip_runtime.h>

#define B_DIM 1024
#define C_DIM 1000
#define H_DIM 768
#define K_DIM 128

typedef __attribute__((ext_vector_type(2))) float v2f;
typedef __attribute__((ext_vector_type(8))) float v8f;

// ---------------------------------------------------------------------------
// Kernel 1: x1 = tanh(relu(-x) @ W_lin^T + b_lin)
//   x  : [B, H] row-major
//   W  : [H, H] row-major (W_lin); B-matrix element (k,n) = W[j0+n, h+k]
//   x1 : [B, H] row-major (workspace)
// One wave (32 lanes) computes a 16x16 output tile via V_WMMA_F32_16X16X4_F32.
// Block = 128 threads = 4 waves stacked along B.
// ---------------------------------------------------------------------------
__global__ void __launch_bounds__(128) fused_gemm_tanh(
    const float* __restrict__ x,
    const float* __restrict__ W,
    const float* __restrict__ b_lin,
    float* __restrict__ x1)
{
    const int lane = threadIdx.x & 31;
    const int wave = threadIdx.x >> 5;
    const int b0 = blockIdx.x * 64 + wave * 16;   // output row base (batch)
    const int j0 = blockIdx.y * 16;               // output col base (hidden)

    // A fragment (16x4 f32): lanes 0-15 hold row M=lane, K = 0,1 in v[0],v[1];
    // lanes 16-31 hold row M=lane-16, K = 2,3.
    const int mrow = lane & 15;
    const int koff = (lane >> 4) * 2;

    const float* xrow = x + (size_t)(b0 + mrow) * H_DIM + koff; // A source
    const float* wrow = W + (size_t)(j0 + mrow) * H_DIM + koff; // B source (n = mrow)

    v8f acc = {};
    for (int h = 0; h < H_DIM; h += 4) {
        v2f a, b;
        // A[m, h+koff .. h+koff+1] = relu(-x)
        a.x = fmaxf(-xrow[h],     0.0f);
        a.y = fmaxf(-xrow[h + 1], 0.0f);
        // B[k, n] = W_lin[j0+n, h+k]  (W^T operand)
        b.x = wrow[h];
        b.y = wrow[h + 1];
        // 8 args: (neg_a, A, neg_b, B, c_mod, C, reuse_a, reuse_b)
        acc = __builtin_amdgcn_wmma_f32_16x16x4_f32(
            false, a, false, b, (short)0, acc, false, false);
    }

    // C/D layout: VGPR r -> M = r + 8*(lane>=16), N = lane&15
    const int n = lane & 15;
    const int mbase = (lane >> 4) * 8;
    const float bj = b_lin[j0 + n];
#pragma unroll
    for (int r = 0; r < 8; ++r) {
        const int m = mbase + r;
        x1[(size_t)(b0 + m) * H_DIM + (j0 + n)] = tanhf(acc[r] + bj);
    }
}

// ---------------------------------------------------------------------------
// Kernel 2: logits[b, k] = sum_h x1[b, h] * class_weights[y[b], h, k]
// One block (128 threads) per sample. Streams the gathered 768x128 f32 slice
// with coalesced float4 loads (~393 KB/block, HBM-bandwidth bound).
//   thread t: column group kg = t & 31 (4 consecutive k), h-slice hs = t >> 5
// ---------------------------------------------------------------------------
__global__ void __launch_bounds__(128) gathered_dot(
    const float* __restrict__ x1,
    const int* __restrict__ y,
    const float* __restrict__ cw,
    float* __restrict__ out)
{
    __shared__ float  sx[H_DIM];
    __shared__ float4 red[4][32];

    const int b = blockIdx.x;
    const int t = threadIdx.x;

    // Stage the x1 row into LDS (3 KB)
    const float* xr = x1 + (size_t)b * H_DIM;
    for (int i = t; i < H_DIM; i += 128) sx[i] = xr[i];
    __syncthreads();

    const int cls = y[b];
    const float4* wp = (const float4*)(cw + (size_t)cls * (H_DIM * K_DIM));

    const int kg = t & 31;   // which group of 4 columns (k = 4*kg .. 4*kg+3)
    const int hs = t >> 5;   // h phase: 0..3

    float4 acc;
    acc.x = 0.0f; acc.y = 0.0f; acc.z = 0.0f; acc.w = 0.0f;

    for (int h = hs; h < H_DIM; h += 4) {
        const float  s = sx[h];
        const float4 w = wp[h * 32 + kg];          // 512 B contiguous per wave
        if (h + 16 < H_DIM)
            __builtin_prefetch(&wp[(h + 16) * 32 + kg], 0, 0);
        acc.x += s * w.x;
        acc.y += s * w.y;
        acc.z += s * w.z;
        acc.w += s * w.w;
    }

    red[hs][kg] = acc;
    __syncthreads();

    if (hs == 0) {
        float4 r0 = red[0][kg], r1 = red[1][kg], r2 = red[2][kg], r3 = red[3][kg];
        float4 o;
        o.x = r0.x + r1.x + r2.x + r3.x;
        o.y = r0.y + r1.y + r2.y + r3.y;
        o.z = r0.z + r1.z + r2.z + r3.z;
        o.w = r0.w + r1.w + r2.w + r3.w;
        ((float4*)out)[(size_t)b * 32 + kg] = o;   // out[b, 4*kg .. 4*kg+3]
    }
}

// ---------------------------------------------------------------------------
// Inputs (setup_inputs order):
//   0: x            [B, H]     f32
//   1: y            [B]        i32
//   2: class_weights[C, H, K]  f32
//   3: class_bias   [C, H]     f32  (unused by the reference math)
//   4: W_lin        [H, H]     f32
//   5: b_lin        [H]        f32
// Output: logits [B, K] f32
// Workspace: x1 [B, H] f32 = 3 MB
// ---------------------------------------------------------------------------
extern "C" void kernel_launch(void* const* d_in, const int* in_sizes, int n_in,
                              void* d_out, int out_size, void* d_ws, size_t ws_size,
                              hipStream_t stream) {
    const float* x    = (const float*)d_in[0];
    const int*   y    = (const int*)d_in[1];
    const float* cw   = (const float*)d_in[2];
    const float* W    = (const float*)d_in[4];
    const float* blin = (const float*)d_in[5];
    float* out = (float*)d_out;
    float* x1  = (float*)d_ws;   // B*H*4 = 3 MB scratch

    dim3 g1(B_DIM / 64, H_DIM / 16);   // 16 x 48 blocks, 128 threads (4 waves)
    fused_gemm_tanh<<<g1, 128, 0, stream>>>(x, W, blin, x1);

    gathered_dot<<<B_DIM, 128, 0, stream>>>(x1, y, cw, out);
}